// TopKSAE_35622458753282
// MI455X (gfx1250) — compile-verified
//
#include <hip/hip_runtime.h>

typedef float v2f __attribute__((ext_vector_type(2)));
typedef float v8f __attribute__((ext_vector_type(8)));

#define D_MODEL 768
#define D_SAE   24576
#define TOPK    32
#define BATCH   4096

// ---------------- Kernel A: z_pre = x @ W_enc + b_enc, zero z_sparse ----------------
#define BM 128
#define BN 128
#define BK 32
#define NCHUNK (D_MODEL / BK)   // 24
#define AS_STRIDE 36    // words per A-row in LDS (conflict-free, 16B-aligned rows)
#define BS_STRIDE 136   // words per B-row in LDS

// Async global->LDS copy of 16 bytes (CDNA5 ASYNCcnt path, no VGPR round trip).
__device__ __forceinline__ void async_copy_b128(unsigned lds_off, const float* gptr) {
    asm volatile("global_load_async_to_lds_b128 %0, %1, off"
                 :: "v"(lds_off), "v"(gptr)
                 : "memory");
}
__device__ __forceinline__ void wait_async0(void) {
    asm volatile("s_wait_asynccnt 0x0" ::: "memory");
}
__device__ __forceinline__ unsigned lds_addr_of(const void* p) {
    // Generic AS pointers to LDS carry the LDS byte address in the low 32 bits.
    return (unsigned)(unsigned long long)p;
}

__global__ __launch_bounds__(256)
void sae_encode_gemm(const float* __restrict__ x,
                     const float* __restrict__ W_enc,
                     const float* __restrict__ b_enc,
                     float* __restrict__ z_pre,
                     float* __restrict__ z_sparse)
{
    __shared__ float As[2][BM * AS_STRIDE];
    __shared__ float Bs[2][BK * BS_STRIDE];

    const int tid   = threadIdx.x;
    const int lane  = tid & 31;
    const int wave  = tid >> 5;            // 0..7
    const int block_m = blockIdx.y * BM;   // grid (D_SAE/BN, BATCH/BM)
    const int block_n = blockIdx.x * BN;

    const int wm = (wave & 3) * 32;        // wave tile: 32(M) x 64(N)
    const int wn = (wave >> 2) * 64;

    const int lhalf = lane >> 4;           // 0 or 1
    const int l15   = lane & 15;

    // Per-thread staging coordinates (4 x b128 for A, 4 x b128 for B per chunk).
    const int a_r0 = tid >> 3;             // 0..31  (rows a_r0, +32, +64, +96)
    const int a_c4 = (tid & 7) * 4;        // 0..28
    const int b_k0 = tid >> 5;             // 0..7   (k rows b_k0, +8, +16, +24)
    const int b_c4 = (tid & 31) * 4;       // 0..124

    v8f acc[2][4];
    #pragma unroll
    for (int i = 0; i < 2; i++)
        #pragma unroll
        for (int j = 0; j < 4; j++)
            #pragma unroll
            for (int r = 0; r < 8; r++) acc[i][j][r] = 0.0f;

    // ---- async stage of one K-chunk into buffer `buf` ----
    auto stage_async = [&](int chunk, int buf) {
        const int kc = chunk * BK;
        #pragma unroll
        for (int rr = 0; rr < BM; rr += 32) {
            const int row = a_r0 + rr;
            async_copy_b128(lds_addr_of(&As[buf][row * AS_STRIDE + a_c4]),
                            &x[(size_t)(block_m + row) * D_MODEL + kc + a_c4]);
        }
        #pragma unroll
        for (int rr = 0; rr < BK; rr += 8) {
            const int kr = b_k0 + rr;
            async_copy_b128(lds_addr_of(&Bs[buf][kr * BS_STRIDE + b_c4]),
                            &W_enc[(size_t)(kc + kr) * D_SAE + block_n + b_c4]);
        }
    };

    // ---- 64 WMMAs on buffer `buf` ----
    auto compute = [&](int buf) {
        const float* asb = As[buf];
        const float* bsb = Bs[buf];
        #pragma unroll
        for (int k4 = 0; k4 < BK; k4 += 4) {
            const int kk = k4 + lhalf * 2;
            // A fragment: 16x4 f32 -> 2 VGPRs (lanes 0-15: K=kk..kk+1 at half 0, 16-31: half 1)
            v2f afrag[2];
            #pragma unroll
            for (int mi = 0; mi < 2; mi++) {
                const float* p = &asb[(wm + mi * 16 + l15) * AS_STRIDE + kk];
                afrag[mi][0] = p[0];
                afrag[mi][1] = p[1];
            }
            // B fragment: 4x16 f32 -> 2 VGPRs: reg r holds row kk+r, col = lane&15
            v2f bfrag[4];
            #pragma unroll
            for (int ni = 0; ni < 4; ni++) {
                const int col = wn + ni * 16 + l15;
                bfrag[ni][0] = bsb[kk * BS_STRIDE + col];
                bfrag[ni][1] = bsb[(kk + 1) * BS_STRIDE + col];
            }
            #pragma unroll
            for (int mi = 0; mi < 2; mi++)
                #pragma unroll
                for (int ni = 0; ni < 4; ni++)
                    acc[mi][ni] = __builtin_amdgcn_wmma_f32_16x16x4_f32(
                        false, afrag[mi], false, bfrag[ni],
                        (short)0, acc[mi][ni], false, false);
        }
    };

    // ---- double-buffered pipeline: one barrier per K-chunk ----
    stage_async(0, 0);
    wait_async0();
    __syncthreads();

    int cur = 0;
    for (int c = 0; c < NCHUNK; ++c) {
        if (c + 1 < NCHUNK) stage_async(c + 1, cur ^ 1);   // overlap with WMMA below
        compute(cur);
        wait_async0();
        __syncthreads();
        cur ^= 1;
    }

    // ---- epilogue: +b_enc, store z_pre, zero z_sparse ----
    #pragma unroll
    for (int mi = 0; mi < 2; mi++) {
        #pragma unroll
        for (int ni = 0; ni < 4; ni++) {
            const int col = block_n + wn + ni * 16 + l15;
            const float be = b_enc[col];
            #pragma unroll
            for (int r = 0; r < 8; r++) {
                const int row = block_m + wm + mi * 16 + r + lhalf * 8;
                const size_t off = (size_t)row * D_SAE + col;
                z_pre[off]    = acc[mi][ni][r] + be;
                z_sparse[off] = 0.0f;
            }
        }
    }
}

// -------- Kernel B: per-row relu -> top-32 -> scatter z_sparse -> decode x_recon --------
// Per-thread chunk: 24 float4s at float4-index {tid + j*256} (coalesced 512B/wave/instr).
__global__ __launch_bounds__(256)
void sae_topk_decode(const float* __restrict__ z_pre,
                     const float* __restrict__ W_dec,
                     const float* __restrict__ b_dec,
                     float* __restrict__ z_sparse,
                     float* __restrict__ x_recon)
{
    extern __shared__ float rowbuf[];      // D_SAE floats (96 KB; WGP has 320 KB LDS)
    __shared__ float wvals[8];
    __shared__ int   widx[8];
    __shared__ float rvals[TOPK];
    __shared__ int   ridx[TOPK];

    const int row = blockIdx.x;
    const int tid = threadIdx.x;
    const int lane = tid & 31;
    const int wave = tid >> 5;
    const size_t rowoff = (size_t)row * D_SAE;

    // load + relu into LDS (b128 path); track per-thread (max, idx) over 24 float4s
    float lmax = -1.0f;
    int   lidx = tid * 4;
    #pragma unroll 4
    for (int j = 0; j < D_SAE / 1024; ++j) {          // 24 iterations
        const int i4 = tid + j * 256;                 // float4 index
        float4 v = *(const float4*)&z_pre[rowoff + (size_t)i4 * 4];
        v.x = v.x > 0.0f ? v.x : 0.0f;
        v.y = v.y > 0.0f ? v.y : 0.0f;
        v.z = v.z > 0.0f ? v.z : 0.0f;
        v.w = v.w > 0.0f ? v.w : 0.0f;
        *(float4*)&rowbuf[i4 * 4] = v;
        const int base = i4 * 4;
        if (v.x > lmax) { lmax = v.x; lidx = base; }
        if (v.y > lmax) { lmax = v.y; lidx = base + 1; }
        if (v.z > lmax) { lmax = v.z; lidx = base + 2; }
        if (v.w > lmax) { lmax = v.w; lidx = base + 3; }
    }
    __syncthreads();

    for (int it = 0; it < TOPK; ++it) {
        // wave32 argmax reduce, tie -> lower index (matches jax top_k ordering)
        float v = lmax; int ix = lidx;
        #pragma unroll
        for (int s = 16; s > 0; s >>= 1) {
            float ov = __shfl_xor(v, s, 32);
            int   oi = __shfl_xor(ix, s, 32);
            if (ov > v || (ov == v && oi < ix)) { v = ov; ix = oi; }
        }
        if (lane == 0) { wvals[wave] = v; widx[wave] = ix; }
        __syncthreads();
        if (tid == 0) {
            float bv = wvals[0]; int bi = widx[0];
            #pragma unroll
            for (int w = 1; w < 8; ++w) {
                if (wvals[w] > bv || (wvals[w] == bv && widx[w] < bi)) {
                    bv = wvals[w]; bi = widx[w];
                }
            }
            rvals[it] = bv; ridx[it] = bi;
        }
        __syncthreads();
        const int win = ridx[it];
        if (((win >> 2) & 255) == tid) {
            // owner removes the winner and rescans only its 24-float4 chunk
            rowbuf[win] = -1.0f;
            float m = -1.0f; int mi = tid * 4;
            for (int j = 0; j < D_SAE / 1024; ++j) {
                const int base = (tid + j * 256) * 4;
                const float4 vv = *(const float4*)&rowbuf[base];
                if (vv.x > m) { m = vv.x; mi = base; }
                if (vv.y > m) { m = vv.y; mi = base + 1; }
                if (vv.z > m) { m = vv.z; mi = base + 2; }
                if (vv.w > m) { m = vv.w; mi = base + 3; }
            }
            lmax = m; lidx = mi;
        }
        __syncthreads();
    }

    // scatter top-k values into (pre-zeroed) z_sparse
    if (tid < TOPK) {
        z_sparse[rowoff + ridx[tid]] = rvals[tid];
    }

    // decode: x_recon[row,:] = sum_k val_k * W_dec[idx_k,:] + b_dec   (W_dec is L2-resident)
    // threads 0..191 each own 4 consecutive columns -> one b128 W_dec load per k
    if (tid < D_MODEL / 4) {
        const int c4 = tid * 4;
        float4 acc = *(const float4*)&b_dec[c4];
        #pragma unroll
        for (int k = 0; k < TOPK; ++k) {
            const float s = rvals[k];
            const float4 w = *(const float4*)&W_dec[(size_t)ridx[k] * D_MODEL + c4];
            acc.x += s * w.x;
            acc.y += s * w.y;
            acc.z += s * w.z;
            acc.w += s * w.w;
        }
        *(float4*)&x_recon[(size_t)row * D_MODEL + c4] = acc;
    }
}

extern "C" void kernel_launch(void* const* d_in, const int* in_sizes, int n_in,
                              void* d_out, int out_size, void* d_ws, size_t ws_size,
                              hipStream_t stream) {
    const float* x     = (const float*)d_in[0];
    const float* W_enc = (const float*)d_in[1];
    const float* b_enc = (const float*)d_in[2];
    const float* W_dec = (const float*)d_in[3];
    const float* b_dec = (const float*)d_in[4];

    float* out      = (float*)d_out;
    float* x_recon  = out;                                   // [BATCH, D_MODEL]
    float* z_sparse = out + (size_t)BATCH * D_MODEL;         // [BATCH, D_SAE]
    float* z_pre    = z_sparse + (size_t)BATCH * D_SAE;      // [BATCH, D_SAE]

    dim3 gridA(D_SAE / BN, BATCH / BM);
    sae_encode_gemm<<<gridA, 256, 0, stream>>>(x, W_enc, b_enc, z_pre, z_sparse);

    sae_topk_decode<<<BATCH, 256, D_SAE * sizeof(float), stream>>>(
        z_pre, W_dec, b_dec, z_sparse, x_recon);
}